// ConditionalVectorQuantizer_48069273977196
// MI455X (gfx1250) — compile-verified
//
#include <hip/hip_runtime.h>
#include <hip/hip_bf16.h>

typedef float v2f __attribute__((ext_vector_type(2)));
typedef float v8f __attribute__((ext_vector_type(8)));

#define NUM_EMB   512
#define EMB_DIM   64
#define N_ROWS    (32 * 64 * 64)          // 131072 flattened vectors
#define Q_SIZE    (32 * 64 * 64 * 64)     // 8388608 quantized elements
#define ENC_SIZE  ((size_t)N_ROWS * NUM_EMB)
#define PERP_OFF  (1 + Q_SIZE)
#define ENC_OFF   (2 + Q_SIZE)
#define LDS_STRIDE 66                     // pad 64 -> 66 to break bank conflicts

// ---------------------------------------------------------------------------
// Kernel 1: zero the one-hot encodings region and the workspace accumulators.
// ---------------------------------------------------------------------------
__global__ void vq_zero(float* __restrict__ enc, float* __restrict__ ws) {
  size_t i      = (size_t)blockIdx.x * blockDim.x + threadIdx.x;
  size_t stride = (size_t)gridDim.x * blockDim.x;
  float2* p = (float2*)enc;                     // enc offset is 8-byte aligned
  const size_t n2 = ENC_SIZE / 2;
  for (size_t k = i; k < n2; k += stride) p[k] = make_float2(0.f, 0.f);
  if (i < 1 + NUM_EMB) ws[i] = 0.f;             // ws[0]=loss acc, ws[1..512]=hist
}

// ---------------------------------------------------------------------------
// Kernel 2: main fused VQ kernel.
//   block = 256 threads = 8 waves; wave w owns rows n0 = blk*128 + w*16.
//   Whole codebook staged in LDS (512 x 66 f32). Distances via f32 WMMA.
// ---------------------------------------------------------------------------
__global__ __launch_bounds__(256)
void vq_main(const float* __restrict__ x,      // inputs, NCHW [32,64,64,64]
             const float* __restrict__ emb,    // [512,64]
             float* __restrict__ q_out,        // quantized, NCHW flat
             float* __restrict__ enc,          // one-hot [N,512]
             float* __restrict__ ws) {         // [0]=loss acc, [1..512]=hist
  extern __shared__ float smem[];
  float* lds_emb = smem;                                  // 512*66
  float* lds_e2  = smem + NUM_EMB * LDS_STRIDE;           // 512
  int*   lds_idx = (int*)(lds_e2 + NUM_EMB);              // 8 waves * 16 rows

  const int tid = threadIdx.x;

  // Stage codebook into LDS (coalesced global reads).
  for (int i = tid; i < NUM_EMB * EMB_DIM; i += 256) {
    int r = i >> 6, c = i & 63;
    lds_emb[r * LDS_STRIDE + c] = emb[i];
  }
  __syncthreads();

  // ||e_k||^2 per code.
  for (int r = tid; r < NUM_EMB; r += 256) {
    float s = 0.f;
    #pragma unroll
    for (int c = 0; c < EMB_DIM; ++c) {
      float v = lds_emb[r * LDS_STRIDE + c];
      s += v * v;
    }
    lds_e2[r] = s;
  }
  __syncthreads();

  const int wid  = tid >> 5;
  const int lane = tid & 31;
  const int half = lane >> 4;   // 0: lanes 0-15, 1: lanes 16-31
  const int l    = lane & 15;

  const int n0 = blockIdx.x * 128 + wid * 16;   // 16 rows per wave, share b,h
  const int w0 = n0 & 63;
  const int h  = (n0 >> 6) & 63;
  const int b  = n0 >> 12;
  // element (b,d,h,w) at ((b*64+d)*64+h)*64 + w ; d-stride = 4096
  const float* xbase = x + ((size_t)b * 4096 + h) * 64;

  // A fragments (ISA 16x4 f32 layout): lanes 0-15 hold K=4kc+{0,1},
  // lanes 16-31 hold K=4kc+{2,3}; M = lane&15 -> row n0+l (w = w0+l).
  v2f af[16];
  float x2l = 0.f;
  #pragma unroll
  for (int kc = 0; kc < 16; ++kc) {
    int d0 = 4 * kc + 2 * half;
    float u = xbase[(size_t)d0 * 4096 + w0 + l];        // coalesced per half
    float v = xbase[(size_t)(d0 + 1) * 4096 + w0 + l];
    af[kc].x = u; af[kc].y = v;
    x2l += u * u + v * v;
  }
  // fold the two K-halves: lanes 0-15 end with ||x_row||^2 for row n0+l
  x2l += __shfl_down(x2l, 16);

  // Running argmin of (||e||^2 - 2 x.e) per C-layout slot.
  float minv[8];
  int   minc[8];
  #pragma unroll
  for (int v = 0; v < 8; ++v) { minv[v] = 3.4e38f; minc[v] = 0; }

  #pragma unroll 1
  for (int j = 0; j < 32; ++j) {                 // 32 column tiles of 16 codes
    v8f acc = {};
    const float* brow = lds_emb + (size_t)(j * 16 + l) * LDS_STRIDE + 2 * half;
    #pragma unroll
    for (int kc = 0; kc < 16; ++kc) {
      v2f bf = *(const v2f*)(brow + 4 * kc);     // B[k, col] = emb[col, k]
      acc = __builtin_amdgcn_wmma_f32_16x16x4_f32(
          false, af[kc], false, bf, (short)0, acc, false, false);
    }
    int   col = j * 16 + l;                      // this lane's column
    float e2c = lds_e2[col];
    #pragma unroll
    for (int v = 0; v < 8; ++v) {
      float d = e2c - 2.f * acc[v];              // ||x||^2 dropped (row-const)
      if (d < minv[v]) { minv[v] = d; minc[v] = col; }
    }
  }

  // Butterfly argmin across the 16 lanes of each half (first-index tiebreak).
  #pragma unroll
  for (int v = 0; v < 8; ++v) {
    #pragma unroll
    for (int off = 8; off >= 1; off >>= 1) {
      float ov = __shfl_xor(minv[v], off);
      int   oc = __shfl_xor(minc[v], off);
      if (ov < minv[v] || (ov == minv[v] && oc < minc[v])) {
        minv[v] = ov; minc[v] = oc;
      }
    }
  }
  // lanes 0-15 now hold result for rows 0-7 (slot v), lanes 16-31 rows 8-15.

  // Distribute per-row argmin so lane l holds idx of row l (via LDS, in-order
  // DS within a wave).
  if (lane == 0) {
    #pragma unroll
    for (int v = 0; v < 8; ++v) lds_idx[wid * 16 + v] = minc[v];
  }
  if (lane == 16) {
    #pragma unroll
    for (int v = 0; v < 8; ++v) lds_idx[wid * 16 + 8 + v] = minc[v];
  }
  int rowidx = lds_idx[wid * 16 + l];

  // One-hot encodings + histogram (one entry per row, lanes 0-15 only).
  if (lane < 16) {
    enc[(size_t)(n0 + l) * NUM_EMB + rowidx] = 1.0f;
    atomicAdd(&ws[1 + rowidx], 1.0f);
  }

  // Loss partial: sum over rows of (minDist + ||x||^2) == sum ||x - e_idx||^2.
  float part = (lane < 16) ? x2l : 0.f;
  {
    float msum = 0.f;
    #pragma unroll
    for (int v = 0; v < 8; ++v) msum += minv[v];
    if (lane == 0 || lane == 16) part += msum;   // each half counted once
  }
  #pragma unroll
  for (int off = 16; off >= 1; off >>= 1) part += __shfl_xor(part, off);
  if (lane == 0) atomicAdd(&ws[0], part);

  // Quantized (straight-through == gathered codes), written back in NCHW.
  // Per c-plane: 16 consecutive w addresses per half-wave -> coalesced.
  float* qbase = q_out + ((size_t)b * 4096 + h) * 64 + w0;
  const float* erow = lds_emb + (size_t)rowidx * LDS_STRIDE;
  #pragma unroll
  for (int c0 = 0; c0 < 32; ++c0) {
    int c = c0 + 32 * half;
    qbase[(size_t)c * 4096 + l] = erow[c];
  }
}

// ---------------------------------------------------------------------------
// Kernel 3: scalars — loss and perplexity.
// ---------------------------------------------------------------------------
__global__ void vq_final(const float* __restrict__ ws, float* __restrict__ out) {
  __shared__ float red[16];
  int t = threadIdx.x;                       // 512 threads, one per code
  float cnt = ws[1 + t];
  float p = cnt / (float)N_ROWS;
  float s = p * __logf(p + 1e-10f);
  #pragma unroll
  for (int off = 16; off >= 1; off >>= 1) s += __shfl_xor(s, off);
  if ((t & 31) == 0) red[t >> 5] = s;
  __syncthreads();
  if (t < 16) {
    float v = red[t];
    #pragma unroll
    for (int off = 8; off >= 1; off >>= 1) v += __shfl_xor(v, off);
    if (t == 0) {
      out[PERP_OFF] = __expf(-v);
      // loss = q_latent + 0.25*e_latent = 1.25 * mean((q-x)^2)
      out[0] = 1.25f * ws[0] / (float)Q_SIZE;
    }
  }
}

// ---------------------------------------------------------------------------
extern "C" void kernel_launch(void* const* d_in, const int* in_sizes, int n_in,
                              void* d_out, int out_size, void* d_ws, size_t ws_size,
                              hipStream_t stream) {
  const float* x   = (const float*)d_in[0];   // inputs [32,64,64,64] f32
  // d_in[1] = labels (unused by reference forward)
  const float* emb = (const float*)d_in[2];   // embedding [512,64] f32
  float* out = (float*)d_out;                 // [loss | quantized | perp | enc]
  float* ws  = (float*)d_ws;

  float* q   = out + 1;
  float* enc = out + ENC_OFF;

  vq_zero<<<4096, 256, 0, stream>>>(enc, ws);

  size_t shmem = (NUM_EMB * LDS_STRIDE + NUM_EMB) * sizeof(float)
               + 8 * 16 * sizeof(int);        // 137728 B — fits 320KB/WGP LDS
  vq_main<<<N_ROWS / 128, 256, shmem, stream>>>(x, emb, q, enc, ws);

  vq_final<<<1, 512, 0, stream>>>(ws, out);
}